// SimpleParametricEQ_26268019982788
// MI455X (gfx1250) — compile-verified
//
#include <hip/hip_runtime.h>
#include <hip/hip_bf16.h>

// SimpleParametricEQ on MI455X (gfx1250, wave32)
// Block-parallel IIR: chunked particular solutions + WMMA-based transition-
// matrix powers + WMMA Kogge-Stone scan over chunk states + correction pass.

#define B_      32
#define C_      2
#define T_      262144
#define K_      6
#define L_      512          // chunk length
#define NCHUNK  512          // T_/L_
#define NLEV    9            // log2(NCHUNK)
#define NSEQ    (B_*C_)      // 64 independent sequences
#define NROWS   (NCHUNK*C_)  // state rows per batch (chunk-major, channel-interleaved)
#define SDIM    16           // 12-dim state padded to 16 for WMMA
#define FS      44100.0f

typedef __attribute__((ext_vector_type(2))) float v2f;   // f32 WMMA A/B frag (16x4 / 4x16)
typedef __attribute__((ext_vector_type(8))) float v8f;   // f32 WMMA C/D frag (16x16)

// ---------------------------------------------------------------- coefficients
__global__ void eq_coeffs_kernel(const float* __restrict__ freq,
                                 const float* __restrict__ gain,
                                 const float* __restrict__ q,
                                 float* __restrict__ cb0, float* __restrict__ cb1,
                                 float* __restrict__ cb2, float* __restrict__ ca1,
                                 float* __restrict__ ca2) {
    int i = blockIdx.x * blockDim.x + threadIdx.x;
    if (i >= B_ * K_) return;
    float A     = __expf(gain[i] * 0.05756462732485114f);   // 10^(g/40) = e^(g*ln10/40)
    float w0    = 6.283185307179586f * freq[i] / FS;
    float cw    = __cosf(w0);
    float alpha = __sinf(w0) / (2.0f * q[i]);
    float b0 = 1.0f + alpha * A;
    float b1 = -2.0f * cw;
    float b2 = 1.0f - alpha * A;
    float a0 = 1.0f + alpha / A;
    float a1 = -2.0f * cw;
    float a2 = 1.0f - alpha / A;
    float inv = 1.0f / a0;
    cb0[i] = b0 * inv; cb1[i] = b1 * inv; cb2[i] = b2 * inv;
    ca1[i] = a1 * inv; ca2[i] = a2 * inv;
}

// -------------------------------------------- one-sample 12x12 transition matrix
// DF2T cascade: per biquad  yk = b0*y + s1 ; s1' = b1*y - a1*yk + s2 ; s2' = b2*y - a2*yk.
// Build the state-to-state map A (x-feedthrough not needed: chunks absorb inputs).
__global__ void build_A_kernel(const float* __restrict__ cb0, const float* __restrict__ cb1,
                               const float* __restrict__ cb2, const float* __restrict__ ca1,
                               const float* __restrict__ ca2, float* __restrict__ Atmp) {
    int b = blockIdx.x * blockDim.x + threadIdx.x;
    if (b >= B_) return;
    float* M = Atmp + b * 256;
    for (int i = 0; i < 256; ++i) M[i] = 0.0f;
    // symbolic vector over [x, s1_0, s2_0, ..., s1_5, s2_5] (13 coeffs)
    float y[13];
    for (int j = 0; j < 13; ++j) y[j] = 0.0f;
    y[0] = 1.0f;
    for (int k = 0; k < K_; ++k) {
        float b0 = cb0[b*K_+k], b1 = cb1[b*K_+k], b2 = cb2[b*K_+k];
        float a1 = ca1[b*K_+k], a2 = ca2[b*K_+k];
        float yk[13], s1r[13], s2r[13];
        for (int j = 0; j < 13; ++j) yk[j] = b0 * y[j];
        yk[1 + 2*k] += 1.0f;                                   // + s1_k
        for (int j = 0; j < 13; ++j) s1r[j] = b1 * y[j] - a1 * yk[j];
        s1r[2 + 2*k] += 1.0f;                                  // + s2_k
        for (int j = 0; j < 13; ++j) s2r[j] = b2 * y[j] - a2 * yk[j];
        for (int j = 0; j < 12; ++j) {
            M[(2*k    ) * 16 + j] = s1r[1 + j];
            M[(2*k + 1) * 16 + j] = s2r[1 + j];
        }
        for (int j = 0; j < 13; ++j) y[j] = yk[j];
    }
}

// ----------------------------- repeated squaring: A^(L*2^m) via v_wmma_f32_16x16x4
// One wave per batch. 16x16x16 product = 4 chained 16x16x4 f32 WMMAs.
__global__ void mat_powers_kernel(const float* __restrict__ Atmp, float* __restrict__ Mpow) {
    __shared__ float sM[256];
    const int b    = blockIdx.x;
    const int lane = threadIdx.x;          // 0..31, wave32
    const int col  = lane & 15;
    const int hi   = lane >> 4;            // 0: K/M 0..1 half, 1: K/M 2..3 half
    for (int i = lane; i < 256; i += 32) sM[i] = Atmp[b * 256 + i];
    __syncthreads();

    for (int sq = 0; sq < 9 + (NLEV - 1); ++sq) {   // after sq=8: A^512 = A^L = M_0
        union { v8f v; float f[8]; } acc;
        for (int v = 0; v < 8; ++v) acc.f[v] = 0.0f;
        #pragma unroll
        for (int kk = 0; kk < 4; ++kk) {
            union { v2f v; float f[2]; } a, bb;
            // A-frag: A[row=col][k], k = 4kk + v + 2*hi
            a.f[0]  = sM[col * 16 + (4*kk + 0 + 2*hi)];
            a.f[1]  = sM[col * 16 + (4*kk + 1 + 2*hi)];
            // B-frag: B[k][n=col] = M[k][col]
            bb.f[0] = sM[(4*kk + 0 + 2*hi) * 16 + col];
            bb.f[1] = sM[(4*kk + 1 + 2*hi) * 16 + col];
            acc.v = __builtin_amdgcn_wmma_f32_16x16x4_f32(
                        false, a.v, false, bb.v, (short)0, acc.v, false, false);
        }
        __syncthreads();   // all frag reads done before overwrite
        for (int v = 0; v < 8; ++v) sM[(v + 8*hi) * 16 + col] = acc.f[v];
        __syncthreads();
        if (sq >= 8) {
            int m = sq - 8;                // m = 0..NLEV-1 : A^(L*2^m)
            float* dst = Mpow + ((size_t)b * NLEV + m) * 256;
            for (int v = 0; v < 8; ++v) dst[(v + 8*hi) * 16 + col] = acc.f[v];
        }
    }
}

// ------------------------------------------------ pass 1: particular end-states
__global__ void pass1_kernel(const float* __restrict__ audio,
                             const float* __restrict__ cb0, const float* __restrict__ cb1,
                             const float* __restrict__ cb2, const float* __restrict__ ca1,
                             const float* __restrict__ ca2, float* __restrict__ P) {
    int gid = blockIdx.x * blockDim.x + threadIdx.x;
    if (gid >= NSEQ * NCHUNK) return;
    int seq = gid / NCHUNK, chunk = gid % NCHUNK;
    int b = seq >> 1, ch = seq & 1;
    float b0[K_], b1[K_], b2[K_], a1[K_], a2[K_], s1[K_], s2[K_];
    #pragma unroll
    for (int k = 0; k < K_; ++k) {
        b0[k] = cb0[b*K_+k]; b1[k] = cb1[b*K_+k]; b2[k] = cb2[b*K_+k];
        a1[k] = ca1[b*K_+k]; a2[k] = ca2[b*K_+k];
        s1[k] = 0.0f; s2[k] = 0.0f;
    }
    const float4* src = (const float4*)(audio + (size_t)seq * T_ + (size_t)chunk * L_);
    for (int t = 0; t < L_ / 4; ++t) {
        float4 x4 = src[t];
        float xs[4] = {x4.x, x4.y, x4.z, x4.w};
        #pragma unroll
        for (int u = 0; u < 4; ++u) {
            float y = xs[u];
            #pragma unroll
            for (int k = 0; k < K_; ++k) {
                float yk = fmaf(b0[k], y, s1[k]);
                s1[k] = fmaf(b1[k], y, fmaf(-a1[k], yk, s2[k]));
                s2[k] = fmaf(b2[k], y, -a2[k] * yk);
                y = yk;
            }
        }
    }
    float* Pr = P + ((size_t)b * NROWS + (size_t)chunk * C_ + ch) * SDIM;
    #pragma unroll
    for (int k = 0; k < K_; ++k) { Pr[2*k] = s1[k]; Pr[2*k + 1] = s2[k]; }
    Pr[12] = Pr[13] = Pr[14] = Pr[15] = 0.0f;
}

// ------------------- Kogge-Stone scan level over chunk states, WMMA combine
// out_j = in_j + A^(L*2^m) * in_{j-2^m}   (16 state rows per wave per WMMA chain)
__global__ void scan_level_kernel(const float* __restrict__ Pin, float* __restrict__ Pout,
                                  const float* __restrict__ Mpow, int m) {
    const int lane = threadIdx.x;                       // wave32, one wave per block
    const int gb   = blockIdx.x;                        // 0 .. B_*NROWS/16 - 1
    const int b    = gb / (NROWS / 16);
    const int r0   = (gb % (NROWS / 16)) * 16;
    const int col  = lane & 15;
    const int hi   = lane >> 4;
    const int shift = (1 << m) * C_;                    // row shift = 2^m chunks
    const float* Pb = Pin + (size_t)b * NROWS * SDIM;
    const float* M  = Mpow + ((size_t)b * NLEV + m) * 256;

    union { v8f v; float f[8]; } acc;                   // C = unshifted rows
    for (int v = 0; v < 8; ++v) acc.f[v] = Pb[(size_t)(r0 + v + 8*hi) * SDIM + col];

    const int prow = r0 + col - shift;                  // A-operand row (shifted state)
    #pragma unroll
    for (int kk = 0; kk < 4; ++kk) {
        union { v2f v; float f[2]; } a, bb;
        for (int v = 0; v < 2; ++v) {
            int k = 4*kk + v + 2*hi;
            a.f[v]  = (prow >= 0) ? Pb[(size_t)prow * SDIM + k] : 0.0f;
            bb.f[v] = M[col * 16 + k];                  // B = M^T : B[k][n=col] = M[col][k]
        }
        acc.v = __builtin_amdgcn_wmma_f32_16x16x4_f32(
                    false, a.v, false, bb.v, (short)0, acc.v, false, false);
    }
    float* Ob = Pout + (size_t)b * NROWS * SDIM;
    for (int v = 0; v < 8; ++v) Ob[(size_t)(r0 + v + 8*hi) * SDIM + col] = acc.f[v];
}

// ------------------------- pass 2: re-filter with correct initial states, emit y
__global__ void pass2_kernel(const float* __restrict__ audio,
                             const float* __restrict__ cb0, const float* __restrict__ cb1,
                             const float* __restrict__ cb2, const float* __restrict__ ca1,
                             const float* __restrict__ ca2, const float* __restrict__ S,
                             float* __restrict__ out) {
    int gid = blockIdx.x * blockDim.x + threadIdx.x;
    if (gid >= NSEQ * NCHUNK) return;
    int seq = gid / NCHUNK, chunk = gid % NCHUNK;
    int b = seq >> 1, ch = seq & 1;
    float b0[K_], b1[K_], b2[K_], a1[K_], a2[K_], s1[K_], s2[K_];
    #pragma unroll
    for (int k = 0; k < K_; ++k) {
        b0[k] = cb0[b*K_+k]; b1[k] = cb1[b*K_+k]; b2[k] = cb2[b*K_+k];
        a1[k] = ca1[b*K_+k]; a2[k] = ca2[b*K_+k];
    }
    if (chunk == 0) {
        #pragma unroll
        for (int k = 0; k < K_; ++k) { s1[k] = 0.0f; s2[k] = 0.0f; }
    } else {
        const float* Sr = S + ((size_t)b * NROWS + (size_t)(chunk - 1) * C_ + ch) * SDIM;
        #pragma unroll
        for (int k = 0; k < K_; ++k) { s1[k] = Sr[2*k]; s2[k] = Sr[2*k + 1]; }
    }
    const float4* src = (const float4*)(audio + (size_t)seq * T_ + (size_t)chunk * L_);
    float4*       dst = (float4*)(out   + (size_t)seq * T_ + (size_t)chunk * L_);
    for (int t = 0; t < L_ / 4; ++t) {
        float4 x4 = src[t];
        float xs[4] = {x4.x, x4.y, x4.z, x4.w};
        #pragma unroll
        for (int u = 0; u < 4; ++u) {
            float y = xs[u];
            #pragma unroll
            for (int k = 0; k < K_; ++k) {
                float yk = fmaf(b0[k], y, s1[k]);
                s1[k] = fmaf(b1[k], y, fmaf(-a1[k], yk, s2[k]));
                s2[k] = fmaf(b2[k], y, -a2[k] * yk);
                y = yk;
            }
            xs[u] = y;
        }
        float4 y4 = {xs[0], xs[1], xs[2], xs[3]};
        dst[t] = y4;
    }
}

// --------------------------------------------------------------------- launcher
extern "C" void kernel_launch(void* const* d_in, const int* in_sizes, int n_in,
                              void* d_out, int out_size, void* d_ws, size_t ws_size,
                              hipStream_t stream) {
    const float* audio = (const float*)d_in[0];   // [32,2,262144]
    const float* freq  = (const float*)d_in[1];   // [32,6]
    const float* gain  = (const float*)d_in[2];   // [32,6]
    const float* qf    = (const float*)d_in[3];   // [32,6]
    float* out = (float*)d_out;

    // workspace layout (floats)
    float* w    = (float*)d_ws;
    float* cb0  = w;                 // 192
    float* cb1  = w + 192;
    float* cb2  = w + 384;
    float* ca1  = w + 576;
    float* ca2  = w + 768;
    float* Atmp = w + 1024;                          // 32*256      =   8192
    float* Mpow = Atmp + (size_t)B_ * 256;           // 32*9*256    =  73728
    float* P0   = Mpow + (size_t)B_ * NLEV * 256;    // 32*1024*16  = 524288
    float* P1   = P0 + (size_t)B_ * NROWS * SDIM;    // same size
    (void)in_sizes; (void)n_in; (void)out_size; (void)ws_size;

    eq_coeffs_kernel<<<1, 256, 0, stream>>>(freq, gain, qf, cb0, cb1, cb2, ca1, ca2);
    build_A_kernel<<<1, 32, 0, stream>>>(cb0, cb1, cb2, ca1, ca2, Atmp);
    mat_powers_kernel<<<B_, 32, 0, stream>>>(Atmp, Mpow);

    const int ntask = NSEQ * NCHUNK;
    pass1_kernel<<<(ntask + 255) / 256, 256, 0, stream>>>(audio, cb0, cb1, cb2, ca1, ca2, P0);

    float* bufs[2] = {P0, P1};
    int cur = 0;
    for (int m = 0; m < NLEV; ++m) {
        scan_level_kernel<<<B_ * (NROWS / 16), 32, 0, stream>>>(bufs[cur], bufs[1 - cur], Mpow, m);
        cur ^= 1;
    }
    pass2_kernel<<<(ntask + 255) / 256, 256, 0, stream>>>(audio, cb0, cb1, cb2, ca1, ca2,
                                                          bufs[cur], out);
}